// RelativePositionalEmbedding_52072183497027
// MI455X (gfx1250) — compile-verified
//
#include <hip/hip_runtime.h>

typedef float v2f __attribute__((ext_vector_type(2)));
typedef float v8f __attribute__((ext_vector_type(8)));

#define B_DIM 2
#define H_DIM 16
#define L_DIM 1024
#define D_DIM 64
#define P_MAX 2046      // last valid pos_emb row (2*1024-2)
#define CT_TOTAL 65     // 65 bands of 16 pos rows cover d in [q0, q0+1039]

// One wave handles one 16-row q-strip of one (b,h) for half of the ct bands.
// Band ct computes T[i,c] = dot(q[q0+i,:], pos_emb[q0+16*ct+c,:]) via 16x
// V_WMMA_F32_16X16X4_F32, then the skew trick stores element (i,c) to
// out[q0+i, (1023-16*ct)+i-c] (predicated). B fragments for band ct+1 are
// loaded into a register ping-pong buffer BEFORE the WMMA chain of band ct,
// so the 16-WMMA chain + stores hide the global-load latency.
__global__ __launch_bounds__(256) void relpos_bias_wmma_f32(
    const float* __restrict__ q,
    const float* __restrict__ pos,
    float* __restrict__ out)
{
    const int lane        = threadIdx.x & 31;
    const int waveInBlock = threadIdx.x >> 5;
    const int gwave       = blockIdx.x * 8 + waveInBlock;   // 0..4095
    const int strip       = gwave >> 1;                     // 0..2047 : (b,h,qtile)
    const int part        = gwave & 1;                      // ct-range split
    const int bh          = strip >> 6;                     // 0..31
    const int qt          = strip & 63;                     // 0..63
    const int q0          = qt << 4;

    const int row  = lane & 15;   // M (for A) / N==c (for B) / N (for C)
    const int half = lane >> 4;   // K-half selector for A/B frags, M-half for C

    // ---- A fragments: a[s] = { q[bh, q0+row, 4s+2*half], q[.., 4s+2*half+1] }
    const float* qbase = q + (size_t)(bh * L_DIM + q0 + row) * D_DIM + 2 * half;
    v2f a[16];
#pragma unroll
    for (int s = 0; s < 16; ++s)
        a[s] = *(const v2f*)(qbase + 4 * s);

    const int ct_begin = part ? 33 : 0;
    const int ct_end   = part ? CT_TOTAL : 33;

    float* obase = out + (size_t)(bh * L_DIM + q0) * L_DIM;

    // B-fragment loader for band ct (dst indices are compile-time constant
    // so dst[] stays in VGPRs).
    auto loadB = [&](int ct, v2f* dst) {
        const int cbase = q0 + (ct << 4);
        int prow = cbase + row;
        if (prow > P_MAX) prow = P_MAX;   // masked-off corner; keep read in-bounds
        const float* pbase = pos + (size_t)prow * D_DIM + 2 * half;
#pragma unroll
        for (int s = 0; s < 16; ++s)
            dst[s] = *(const v2f*)(pbase + 4 * s);
    };

    // 16-WMMA chain + skewed predicated stores for band ct.
    auto computeStore = [&](int ct, const v2f* b) {
        v8f acc = {};
#pragma unroll
        for (int s = 0; s < 16; ++s)
            acc = __builtin_amdgcn_wmma_f32_16x16x4_f32(
                /*neg_a=*/false, a[s], /*neg_b=*/false, b[s],
                /*c_mod=*/(short)0, acc, /*reuse_a=*/false, /*reuse_b=*/false);

        const int K0 = 1023 - (ct << 4);
#pragma unroll
        for (int v = 0; v < 8; ++v) {
            const int i = v + 8 * half;       // C layout: VGPR v, lane-half -> M
            const int k = K0 + i - row;       // row = c = N
            if ((unsigned)k < (unsigned)L_DIM)
                obase[(size_t)i * L_DIM + k] = acc[v] * 0.125f;
        }
    };

    // ---- software pipeline with register ping-pong (b0 / b1) ----
    v2f b0[16], b1[16];
    int ct = ct_begin;
    loadB(ct, b0);
    for (;;) {
        // even phase: compute on b0, prefetch ct+1 into b1
        if (ct + 1 < ct_end) {
            loadB(ct + 1, b1);
            computeStore(ct, b0);
            ++ct;
        } else {
            computeStore(ct, b0);
            break;
        }
        // odd phase: compute on b1, prefetch ct+1 into b0
        if (ct + 1 < ct_end) {
            loadB(ct + 1, b0);
            computeStore(ct, b1);
            ++ct;
        } else {
            computeStore(ct, b1);
            break;
        }
    }
}

extern "C" void kernel_launch(void* const* d_in, const int* in_sizes, int n_in,
                              void* d_out, int out_size, void* d_ws, size_t ws_size,
                              hipStream_t stream) {
    const float* q   = (const float*)d_in[0];   // [2,16,1024,64] f32
    // d_in[1] is k (unused by the reference math: bias depends only on q, pos_emb)
    const float* pos = (const float*)d_in[2];   // [2047,64] f32
    float* out = (float*)d_out;                 // [2,16,1024,1024] f32

    // 2048 strips * 2 ct-parts = 4096 waves; 8 waves (256 thr) per block.
    relpos_bias_wmma_f32<<<512, 256, 0, stream>>>(q, pos, out);
}